// LightGCN_59468117180654
// MI455X (gfx1250) — compile-verified
//
#include <hip/hip_runtime.h>

// ---------------------------------------------------------------------------
// LightGCN propagation for MI455X (gfx1250).
//   E = [user_emb; item_emb]  (150000 x 128 f32, 76.8 MB)
//   3x: next = A_coo * cur (gather*w -> scatter-add), acc += next
//   out = acc / 4
// Memory/atomic bound (AI < 1 FLOP/B): edge stream (192 MB over 3 layers,
// loaded NT so it never evicts L2-resident embeddings); cur/next (~154 MB)
// stay hot in the 192 MB L2 for the random gathers + f32 scatter atomics.
// CDNA5 paths: double-buffered async global->LDS edge staging (ASYNCcnt),
// global_prefetch for gathers, hw global_atomic_add_f32, wave32 row shape
// (32 lanes x float4 = 128 = DIM).
// ---------------------------------------------------------------------------

#define DIM 128
#define CHUNK 1024          // edges staged to LDS per pipeline stage
#define SCATTER_BLOCK 256   // 8 waves

typedef __attribute__((ext_vector_type(4))) int v4i;
typedef __attribute__((address_space(1))) v4i* gv4i_p;   // global
typedef __attribute__((address_space(3))) v4i* lv4i_p;   // LDS

// ---- CDNA5 async global->LDS copy (16 B per lane), NT temporal hint -------
__device__ __forceinline__ void async_b128_nt(void* lds_dst, const void* gsrc) {
#if __has_builtin(__builtin_amdgcn_global_load_async_to_lds_b128)
  // cpol: TH[2:0]=1 (non-temporal) -> don't pollute L2 with the edge stream
  __builtin_amdgcn_global_load_async_to_lds_b128(
      (gv4i_p)gsrc, (lv4i_p)lds_dst, 0, 1);
#else
  unsigned lds_off =
      (unsigned)(unsigned long long)(__attribute__((address_space(3))) char*)lds_dst;
  asm volatile("global_load_async_to_lds_b128 %0, %1, off"
               :: "v"(lds_off), "v"(gsrc) : "memory");
#endif
}

__device__ __forceinline__ void wait_async0() {
#if __has_builtin(__builtin_amdgcn_s_wait_asynccnt)
  __builtin_amdgcn_s_wait_asynccnt(0);
#else
  asm volatile("s_wait_asynccnt 0" ::: "memory");
#endif
}

__device__ __forceinline__ void wait_async3() {
#if __has_builtin(__builtin_amdgcn_s_wait_asynccnt)
  __builtin_amdgcn_s_wait_asynccnt(3);   // 3 next-chunk ops may stay in flight
#else
  asm volatile("s_wait_asynccnt 3" ::: "memory");
#endif
}

// ---- f32 scatter-add: hardware global_atomic_add_f32 (L2 atomic, no ret) --
__device__ __forceinline__ void atomic_add_f32(float* p, float v) {
#if defined(__HIP_DEVICE_COMPILE__)
  unsafeAtomicAdd(p, v);
#else
  atomicAdd(p, v);
#endif
}

// ---------------------------------------------------------------------------
// init: cur = acc = [ue; ie], next = 0          (one float4 per thread)
// ---------------------------------------------------------------------------
__global__ void lightgcn_init(const float4* __restrict__ ue,
                              const float4* __restrict__ ie,
                              float4* __restrict__ cur,
                              float4* __restrict__ nxt,
                              float4* __restrict__ acc,
                              int nu4, int n4) {
  int i = blockIdx.x * blockDim.x + threadIdx.x;
  if (i >= n4) return;
  const float4 v = (i < nu4) ? ue[i] : ie[i - nu4];
  cur[i] = v;
  acc[i] = v;
  nxt[i] = make_float4(0.f, 0.f, 0.f, 0.f);
}

// ---------------------------------------------------------------------------
// addzero: acc = (acc + nxt) * scale ; zbuf = 0  (zbuf = next layer's "next")
// ---------------------------------------------------------------------------
__global__ void lightgcn_addzero(float4* __restrict__ acc,
                                 const float4* __restrict__ nxt,
                                 float4* __restrict__ zbuf,
                                 float scale, int n4) {
  int i = blockIdx.x * blockDim.x + threadIdx.x;
  if (i >= n4) return;
  float4 a = acc[i];
  float4 b = nxt[i];
  a.x = (a.x + b.x) * scale;
  a.y = (a.y + b.y) * scale;
  a.z = (a.z + b.z) * scale;
  a.w = (a.w + b.w) * scale;
  acc[i] = a;
  zbuf[i] = make_float4(0.f, 0.f, 0.f, 0.f);
}

// ---------------------------------------------------------------------------
// scatter: for each edge e: nxt[row[e]] += cur[col[e]] * w[e]
// Persistent blocks; edge metadata double-buffered in LDS via async DMA so
// the next chunk streams in (ASYNCcnt<=3) while the current one is processed.
// One wave per edge; lane handles 4 consecutive dwords (32*4 = 128 = DIM).
// ---------------------------------------------------------------------------
__global__ void __launch_bounds__(SCATTER_BLOCK)
lightgcn_scatter(const int* __restrict__ rows,
                 const int* __restrict__ cols,
                 const float* __restrict__ ew,
                 const float* __restrict__ cur,
                 float* __restrict__ nxt,
                 int nnz) {
  __shared__ int   s_row[2][CHUNK];
  __shared__ int   s_col[2][CHUNK];
  __shared__ float s_w[2][CHUNK];     // 24 KB of the 320 KB WGP LDS

  const int tid    = threadIdx.x;
  const int lane   = tid & 31;
  const int wv     = tid >> 5;                      // 8 waves / block
  const int stride = gridDim.x;
  const int nchunks = (nnz + CHUNK - 1) / CHUNK;

  // Stage chunk c into LDS buffer b. Full chunks (block-uniform condition)
  // use exactly 3 async ops per wave; the single ragged tail chunk (which
  // never has a successor) uses plain loads -> per-wave ASYNCcnt stays exact.
  auto stage = [&](int c, int b) -> bool {
    const int base = c * CHUNK;
    const int cnt  = min(CHUNK, nnz - base);
    const int i0   = tid * 4;                       // 256 thr * 4 = 1024 edges
    if (cnt == CHUNK) {
      async_b128_nt(&s_row[b][i0], rows + base + i0);
      async_b128_nt(&s_col[b][i0], cols + base + i0);
      async_b128_nt(&s_w[b][i0],   ew   + base + i0);
      return true;
    }
    for (int k = 0; k < 4; ++k) {
      const int i = i0 + k;
      if (i < cnt) {
        s_row[b][i] = rows[base + i];
        s_col[b][i] = cols[base + i];
        s_w[b][i]   = ew[base + i];
      }
    }
    return false;
  };

  int c = blockIdx.x;
  if (c >= nchunks) return;                         // uniform per block
  int p = 0;
  stage(c, p);                                      // prime the pipeline

  for (; c < nchunks; c += stride) {
    const int cn = c + stride;
    bool next_async = false;
    if (cn < nchunks) next_async = stage(cn, p ^ 1);

    if (next_async) wait_async3();                  // current chunk landed,
    else            wait_async0();                  // next stays in flight
    __syncthreads();

    const int cnt = min(CHUNK, nnz - c * CHUNK);
    for (int j = wv; j < cnt; j += SCATTER_BLOCK / 32) {
      const int jn = j + SCATTER_BLOCK / 32;
      if (jn < cnt) {
        // hide the random L2 gather latency of this wave's next edge
        __builtin_prefetch(cur + (size_t)s_col[p][jn] * DIM + lane * 4, 0, 3);
      }
      const int   r  = s_row[p][j];                 // LDS broadcast reads
      const int   ci = s_col[p][j];
      const float w  = s_w[p][j];

      const float4 v = *(const float4*)(cur + (size_t)ci * DIM + lane * 4);
      float* dst = nxt + (size_t)r * DIM + lane * 4;
      atomic_add_f32(dst + 0, v.x * w);
      atomic_add_f32(dst + 1, v.y * w);
      atomic_add_f32(dst + 2, v.z * w);
      atomic_add_f32(dst + 3, v.w * w);
    }
    __syncthreads();                                // LDS buffer reuse fence
    p ^= 1;
  }
}

// ---------------------------------------------------------------------------
extern "C" void kernel_launch(void* const* d_in, const int* in_sizes, int n_in,
                              void* d_out, int out_size, void* d_ws, size_t ws_size,
                              hipStream_t stream) {
  const float* ue = (const float*)d_in[0];          // [nu, 128] f32
  const float* ie = (const float*)d_in[1];          // [ni, 128] f32
  const int*   ei = (const int*)d_in[2];            // [2, nnz] int32 (per harness int mapping)
  const float* ew = (const float*)d_in[3];          // [nnz] f32

  const int nu  = in_sizes[0] / DIM;
  const int ni  = in_sizes[1] / DIM;
  const int nnz = in_sizes[3];
  const int nnodes = nu + ni;

  const long elems = (long)nnodes * DIM;            // 19.2M floats
  const int  n4    = (int)(elems / 4);
  const int  nu4   = (nu * DIM) / 4;

  float* cur = (float*)d_ws;                        // 76.8 MB
  float* nxt = cur + elems;                         // 76.8 MB
  float* acc = (float*)d_out;                       // output doubles as acc

  const int tpb = 256;
  const int gb  = (n4 + tpb - 1) / tpb;

  lightgcn_init<<<gb, tpb, 0, stream>>>(
      (const float4*)ue, (const float4*)ie,
      (float4*)cur, (float4*)nxt, (float4*)acc, nu4, n4);

  const int* rows  = ei;
  const int* colsp = ei + nnz;
  const int nchunks = (nnz + CHUNK - 1) / CHUNK;
  const int nblocks = (nchunks < 1024) ? nchunks : 1024;  // persistent grid

  for (int L = 0; L < 3; ++L) {
    lightgcn_scatter<<<nblocks, SCATTER_BLOCK, 0, stream>>>(
        rows, colsp, ew, cur, nxt, nnz);
    const float s = (L == 2) ? 0.25f : 1.0f;
    lightgcn_addzero<<<gb, tpb, 0, stream>>>(
        (float4*)acc, (const float4*)nxt, (float4*)cur, s, n4);
    float* t = cur; cur = nxt; nxt = t;             // swap roles
  }
}